// AGCN_79783312490660
// MI455X (gfx1250) — compile-verified
//
#include <hip/hip_runtime.h>

typedef __attribute__((ext_vector_type(2))) float        v2f;
typedef __attribute__((ext_vector_type(8))) float        v8f;
typedef __attribute__((ext_vector_type(4))) unsigned int v4u;
typedef __attribute__((ext_vector_type(8))) int          v8i;
typedef __attribute__((ext_vector_type(4))) int          v4i;

#define N_NODES 4000
#define BATCH   64
#define CIN     64
#define COUT    64
#define DEMB    10
#define CHEB    3

// ---------------------------------------------------------------------------
// TDM helper: issue a 2D tile load (global -> LDS) with LDS row padding.
//   tile_d0 x tile_d1 elements (f32), row stride stride0 elements in memory.
//   pad_interval: log2-style encoding per D# spec (4 -> every 32 DWORDs)
//   pad_amount:   (DWORDs of pad) - 1
// ---------------------------------------------------------------------------
__device__ __forceinline__ void tdm_load_2d(
    const void* gaddr, unsigned lds_byte_off,
    unsigned tile_d0, unsigned tile_d1,
    unsigned tensor_d0, unsigned tensor_d1,
    unsigned stride0,
    unsigned pad_interval, unsigned pad_amount)
{
    const unsigned long long ga = (unsigned long long)(uintptr_t)gaddr;
    v4u g0;
    g0.x = 1u;                                         // count=1, is_restore=0
    g0.y = lds_byte_off;                               // lds_addr [63:32]
    g0.z = (unsigned)(ga & 0xffffffffu);               // global_addr lo
    g0.w = (unsigned)((ga >> 32) & 0x01ffffffu)        // global_addr hi (57b)
         | (2u << 30);                                 // type = 2 (image)
    v8i g1;
    g1[0] = (int)((2u << 16)                           // data_size = 4B
                | (1u << 20)                           // pad_enable
                | (pad_interval << 22)
                | (pad_amount << 25));
    g1[1] = (int)((tensor_d0 & 0xffffu) << 16);        // tensor_dim0 lo16
    g1[2] = (int)(((tensor_d0 >> 16) & 0xffffu)        // tensor_dim0 hi16
                | ((tensor_d1 & 0xffffu) << 16));      // tensor_dim1 lo16
    g1[3] = (int)(((tensor_d1 >> 16) & 0xffffu)        // tensor_dim1 hi16
                | ((tile_d0 & 0xffffu) << 16));        // tile_dim0
    g1[4] = (int)(tile_d1 & 0xffffu);                  // tile_dim1 (tile_dim2=0)
    g1[5] = (int)stride0;                              // tensor_dim0_stride lo32
    g1[6] = 0;                                         // stride hi / dim1_stride lo
    g1[7] = 0;
    const v4i z4 = {0, 0, 0, 0};                       // 2D: groups 2/3 unused
    const v8i z8 = {0, 0, 0, 0, 0, 0, 0, 0};
    __builtin_amdgcn_tensor_load_to_lds(g0, g1, z4, z4, z8, 0);
}

// ---------------------------------------------------------------------------
// Kernel 1: A[n,m] = softmax_m( relu(e_n . e_m) ), one block per row n.
// ---------------------------------------------------------------------------
__global__ __launch_bounds__(256) void adj_softmax_kernel(
    const float* __restrict__ emb, float* __restrict__ A)
{
    __shared__ float buf[N_NODES];
    __shared__ float red[256];
    __shared__ float en[DEMB];
    const int n   = blockIdx.x;
    const int tid = threadIdx.x;
    if (tid < DEMB) en[tid] = emb[n * DEMB + tid];
    __syncthreads();

    float lmax = -1e30f;
    for (int m = tid; m < N_NODES; m += 256) {
        const float* em = emb + m * DEMB;
        float s = 0.f;
#pragma unroll
        for (int d = 0; d < DEMB; ++d) s += en[d] * em[d];
        s = s > 0.f ? s : 0.f;       // relu
        buf[m] = s;
        lmax = s > lmax ? s : lmax;
    }
    red[tid] = lmax; __syncthreads();
    for (int off = 128; off > 0; off >>= 1) {
        if (tid < off) red[tid] = fmaxf(red[tid], red[tid + off]);
        __syncthreads();
    }
    const float rmax = red[0];
    __syncthreads();

    float lsum = 0.f;
    for (int m = tid; m < N_NODES; m += 256) {
        float v = __expf(buf[m] - rmax);
        buf[m] = v;
        lsum += v;
    }
    red[tid] = lsum; __syncthreads();
    for (int off = 128; off > 0; off >>= 1) {
        if (tid < off) red[tid] += red[tid + off];
        __syncthreads();
    }
    const float inv = 1.f / red[0];
    for (int m = tid; m < N_NODES; m += 256)
        A[(size_t)n * N_NODES + m] = buf[m] * inv;
}

// ---------------------------------------------------------------------------
// Kernel 2: Y[b,n,c] = alpha * sum_m A[n,m] * X[b,m,c] (+ beta * Xadd[b,n,c])
//
// Block tile: 32 node-rows x 4 batches. 8 waves: wave = (wr, wb); each wave
// owns 16 rows x 64 cols of batch b0+wb as four 16x16 f32 WMMA accumulators
// (one A-fragment reused across 4 B-fragments per K-step).
//
// Staging: Tensor Data Mover, double-buffered. Wave 0 issues 5 TDM
// descriptors (A tile + 4 X tiles) for chunk i+1 while all waves compute
// chunk i; TENSORcnt is in-order per wave, so s_wait_tensorcnt(5) after the
// next-chunk issue guarantees the current chunk's 5 descriptors completed.
// TDM pad_enable reproduces the bank-padding (A: stride 34, X: stride 66).
// ---------------------------------------------------------------------------
template <bool USE_BETA>
__global__ __launch_bounds__(256) void spmm_kernel(
    const float* __restrict__ A,
    const float* __restrict__ X,
    const float* __restrict__ Xadd,
    float* __restrict__ Y,
    float alpha, float beta)
{
    constexpr int TM   = 32;            // node rows per block
    constexpr int NB   = 4;             // batches per block
    constexpr int KC   = 32;            // k chunk (reduction over m)
    constexpr int LDA  = KC + 2;        // 34: A LDS row stride (pad 2/32)
    constexpr int LDX  = CIN + 2;       // 66: X LDS row stride (pad 2/64)
    constexpr int ABUF = TM * LDA;      // 1088 floats
    constexpr int XB1  = KC * LDX;      // 2112 floats per batch tile
    constexpr int XBUF = NB * XB1;      // 8448 floats

    // single arena => known LDS byte offsets (group segment starts at 0)
    __shared__ __align__(16) float smem[2 * ABUF + 2 * XBUF];

    const int nBase = blockIdx.x * TM;
    const int b0    = blockIdx.y * NB;
    const int tid   = threadIdx.x;
    const int wave  = tid >> 5;
    const int lane  = tid & 31;
    const int half  = lane >> 4;        // 0: K=0,1  1: K=2,3 (16x4 A layout)
    const int l     = lane & 15;
    const int wr    = wave >> 2;        // row group 0..1 (16 rows each)
    const int wb    = wave & 3;         // batch within block

    v8f acc0 = {0.f, 0.f, 0.f, 0.f, 0.f, 0.f, 0.f, 0.f};
    v8f acc1 = acc0, acc2 = acc0, acc3 = acc0;

    auto issue_chunk = [&](int pb, int m0) {
        // A tile: 32 rows (stride N_NODES) x 32 cols -> LDS [row][34]
        tdm_load_2d(&A[(size_t)nBase * N_NODES + m0],
                    (unsigned)(pb * ABUF) * 4u,
                    /*tile*/ KC, TM, /*tensor*/ N_NODES, N_NODES,
                    /*stride0*/ N_NODES, /*pad: every 32 dw*/ 4u, /*+2 dw*/ 1u);
        // X tiles: 32 k-rows (stride CIN) x 64 cols -> LDS [k][66]
#pragma unroll
        for (int q = 0; q < NB; ++q)
            tdm_load_2d(&X[((size_t)(b0 + q) * N_NODES + m0) * CIN],
                        (unsigned)(2 * ABUF + pb * XBUF + q * XB1) * 4u,
                        /*tile*/ CIN, KC, /*tensor*/ CIN, N_NODES,
                        /*stride0*/ CIN, /*pad: every 64 dw*/ 5u, /*+2 dw*/ 1u);
    };

    if (wave == 0) issue_chunk(0, 0);

    int pb = 0;
    for (int m0 = 0; m0 < N_NODES; m0 += KC) {
        const bool hasNext = (m0 + KC) < N_NODES;
        if (wave == 0) {
            if (hasNext) {
                issue_chunk(pb ^ 1, m0 + KC);
                __builtin_amdgcn_s_wait_tensorcnt(5);  // current chunk done
            } else {
                __builtin_amdgcn_s_wait_tensorcnt(0);
            }
        }
        __syncthreads();   // publish current buffer to all waves

        const float* sAc = smem + pb * ABUF;
        const float* sXc = smem + 2 * ABUF + pb * XBUF + wb * XB1;
#pragma unroll
        for (int k0 = 0; k0 < KC; k0 += 4) {
            const int ks = k0 + 2 * half;
            v2f a  = *(const v2f*)(sAc + (16 * wr + l) * LDA + ks);
            // B frags from row-major [k][66]: 2 dwords 264B apart
            //   -> single ds_load_2addr_b32 each
            v2f f0 = { sXc[ks * LDX + l     ], sXc[(ks + 1) * LDX + l     ] };
            v2f f1 = { sXc[ks * LDX + l + 16], sXc[(ks + 1) * LDX + l + 16] };
            v2f f2 = { sXc[ks * LDX + l + 32], sXc[(ks + 1) * LDX + l + 32] };
            v2f f3 = { sXc[ks * LDX + l + 48], sXc[(ks + 1) * LDX + l + 48] };
            acc0 = __builtin_amdgcn_wmma_f32_16x16x4_f32(false, a, false, f0, (short)0, acc0, false, false);
            acc1 = __builtin_amdgcn_wmma_f32_16x16x4_f32(false, a, false, f1, (short)0, acc1, false, false);
            acc2 = __builtin_amdgcn_wmma_f32_16x16x4_f32(false, a, false, f2, (short)0, acc2, false, false);
            acc3 = __builtin_amdgcn_wmma_f32_16x16x4_f32(false, a, false, f3, (short)0, acc3, false, false);
        }
        __syncthreads();   // all reads done before TDM reuses this buffer
        pb ^= 1;
    }

    // epilogue: C/D layout -> VGPR j holds (M = j + 8*half, N = l)
    const size_t bOff = (size_t)(b0 + wb) * N_NODES * CIN;
    float*       Yb   = Y + bOff;
    const float* Ab   = USE_BETA ? (Xadd + bOff) : nullptr;
#pragma unroll
    for (int j = 0; j < 8; ++j) {
        const int row = nBase + 16 * wr + j + 8 * half;
        float v0 = alpha * acc0[j];
        float v1 = alpha * acc1[j];
        float v2 = alpha * acc2[j];
        float v3 = alpha * acc3[j];
        if (USE_BETA) {
            v0 += beta * Ab[(size_t)row * CIN + l];
            v1 += beta * Ab[(size_t)row * CIN + l + 16];
            v2 += beta * Ab[(size_t)row * CIN + l + 32];
            v3 += beta * Ab[(size_t)row * CIN + l + 48];
        }
        Yb[(size_t)row * CIN + l     ] = v0;
        Yb[(size_t)row * CIN + l + 16] = v1;
        Yb[(size_t)row * CIN + l + 32] = v2;
        Yb[(size_t)row * CIN + l + 48] = v3;
    }
}

// ---------------------------------------------------------------------------
// Kernel 3: one block per node n. Fuses per-node weight/bias generation with
// the 64x192x64 output GEMM (WMMA f32 16x16x4; 8 waves x 2 accumulators).
// ---------------------------------------------------------------------------
__global__ __launch_bounds__(256) void out_kernel(
    const float* __restrict__ x,
    const float* __restrict__ Y1,
    const float* __restrict__ Y2,
    const float* __restrict__ emb,
    const float* __restrict__ wpool,
    const float* __restrict__ bpool,
    float* __restrict__ out)
{
    constexpr int KP = CHEB * CIN;   // 192
    constexpr int LD = KP + 2;       // even pad
    __shared__ __align__(16) float sXg[BATCH][LD];  // A matrix [b][p]
    __shared__ __align__(16) float sWt[COUT][LD];   // B matrix transposed [o][p]
    __shared__ float sBias[COUT];
    __shared__ float en[DEMB];

    const int n   = blockIdx.x;
    const int tid = threadIdx.x;
    if (tid < DEMB) en[tid] = emb[n * DEMB + tid];
    __syncthreads();

    for (int idx = tid; idx < KP * COUT; idx += 256) {
        const int p = idx >> 6;
        const int o = idx & 63;
        float s = 0.f;
#pragma unroll
        for (int d = 0; d < DEMB; ++d)
            s += en[d] * wpool[((size_t)d * KP + p) * COUT + o];
        sWt[o][p] = s;
    }
    if (tid < COUT) {
        float s = 0.f;
#pragma unroll
        for (int d = 0; d < DEMB; ++d) s += en[d] * bpool[d * COUT + tid];
        sBias[tid] = s;
    }
    for (int idx = tid; idx < BATCH * KP; idx += 256) {
        const int bb = idx / KP;
        const int p  = idx - bb * KP;
        const int k  = p >> 6;
        const int c  = p & 63;
        const float* src = (k == 0) ? x : (k == 1 ? Y1 : Y2);
        sXg[bb][p] = src[((size_t)bb * N_NODES + n) * CIN + c];
    }
    __syncthreads();

    const int wave = tid >> 5;
    const int lane = tid & 31;
    const int half = lane >> 4;
    const int l    = lane & 15;
    const int wr   = wave & 3;    // batch-row group 0..3 (16 rows each)
    const int wcg  = wave >> 2;   // 0..1 -> cols [0,32) or [32,64)

    v8f acc0 = {0.f, 0.f, 0.f, 0.f, 0.f, 0.f, 0.f, 0.f};
    v8f acc1 = acc0;
#pragma unroll 4
    for (int k0 = 0; k0 < KP; k0 += 4) {
        const int ks = k0 + 2 * half;
        v2f a  = *(const v2f*)&sXg[16 * wr + l][ks];
        v2f b0 = *(const v2f*)&sWt[32 * wcg + l][ks];
        v2f b1 = *(const v2f*)&sWt[32 * wcg + 16 + l][ks];
        acc0 = __builtin_amdgcn_wmma_f32_16x16x4_f32(
            false, a, false, b0, (short)0, acc0, false, false);
        acc1 = __builtin_amdgcn_wmma_f32_16x16x4_f32(
            false, a, false, b1, (short)0, acc1, false, false);
    }

#pragma unroll
    for (int j = 0; j < 8; ++j) {
        const int bb = 16 * wr + j + 8 * half;
        const int o0 = 32 * wcg + l;
        const int o1 = o0 + 16;
        out[((size_t)bb * N_NODES + n) * COUT + o0] = acc0[j] + sBias[o0];
        out[((size_t)bb * N_NODES + n) * COUT + o1] = acc1[j] + sBias[o1];
    }
}

// ---------------------------------------------------------------------------
extern "C" void kernel_launch(void* const* d_in, const int* in_sizes, int n_in,
                              void* d_out, int out_size, void* d_ws, size_t ws_size,
                              hipStream_t stream)
{
    const float* x     = (const float*)d_in[0];  // [64,4000,64]
    const float* emb   = (const float*)d_in[1];  // [4000,10]
    const float* wpool = (const float*)d_in[2];  // [10,3,64,64]
    const float* bpool = (const float*)d_in[3];  // [10,64]
    float* out = (float*)d_out;                  // [64,4000,64]

    // workspace layout: A (64 MB) | Y1 (65.5 MB) | Y2 (65.5 MB)
    float* A  = (float*)d_ws;
    float* Y1 = A  + (size_t)N_NODES * N_NODES;
    float* Y2 = Y1 + (size_t)BATCH * N_NODES * CIN;

    adj_softmax_kernel<<<N_NODES, 256, 0, stream>>>(emb, A);

    dim3 g(N_NODES / 32, BATCH / 4);
    // Y1 = A @ X           (Chebyshev T1 term)
    spmm_kernel<false><<<g, 256, 0, stream>>>(A, x, nullptr, Y1, 1.0f, 0.0f);
    // Y2 = 2*A @ Y1 - X    (Chebyshev T2 term, avoids materializing A^2)
    spmm_kernel<true ><<<g, 256, 0, stream>>>(A, Y1, x, Y2, 2.0f, -1.0f);

    out_kernel<<<N_NODES, 256, 0, stream>>>(x, Y1, Y2, emb, wpool, bpool, out);
}